// Normalizer_15032385536167
// MI455X (gfx1250) — compile-verified
//
#include <hip/hip_runtime.h>
#include <hip/hip_bf16.h>
#include <stdint.h>

// Row-wise normalize: out[r, c] = in[r, c] * inv_deg(r), inv_deg = 1/rowsum
// (non-finite -> 0).  2*32*1024 = 65536 rows of 1024 f32.
//
// Pure HBM-bandwidth problem (512 MB traffic, ~22 us floor at 23.3 TB/s).
// One 256-thread block (8 wave32) per 4 KB row:
//   global --(GLOBAL_LOAD_ASYNC_TO_LDS_B128, ASYNCcnt)--> LDS
//   LDS --(ds_load_b128)--> VGPR, wave32 ds_swizzle butterfly reduce +
//   LDS cross-wave combine, scale in registers, NT global_store_b128 out.

#define NCOLS 1024
#define TPB   256
#define NWAVES (TPB / 32)

// Native clang vector type: required by __builtin_nontemporal_store and gives
// clean b128 codegen.
typedef float v4f __attribute__((ext_vector_type(4)));

// ds_swizzle group-of-32 XOR pattern: offset = xor_mask<<10 | or<<5 | and(0x1f)
__device__ __forceinline__ float swizzle_xor_add(float x, int pattern) {
    union { float f; int i; } u;
    u.f = x;
    switch (pattern) {  // ds_swizzle offset must be an immediate
        case 16: u.i = __builtin_amdgcn_ds_swizzle(u.i, 0x401f); break; // SWAPX16
        case 8:  u.i = __builtin_amdgcn_ds_swizzle(u.i, 0x201f); break; // SWAPX8
        case 4:  u.i = __builtin_amdgcn_ds_swizzle(u.i, 0x101f); break; // SWAPX4
        case 2:  u.i = __builtin_amdgcn_ds_swizzle(u.i, 0x081f); break; // SWAPX2
        default: u.i = __builtin_amdgcn_ds_swizzle(u.i, 0x041f); break; // SWAPX1
    }
    return x + u.f;
}

__global__ __launch_bounds__(TPB) void rownorm_kernel(
    const float* __restrict__ in, float* __restrict__ out)
{
    __shared__ float row[NCOLS];        // one 4 KB row staged in LDS
    __shared__ float wsum[NWAVES];      // per-wave partial sums

    const int t = threadIdx.x;
    const size_t rowBase = (size_t)blockIdx.x * (size_t)NCOLS;
    const float* gsrc = in + rowBase;           // uniform per block -> SGPR pair
    const unsigned voff = (unsigned)t * 16u;    // byte offset of this lane's 16B

    // LDS byte address for this lane's 16-byte chunk. Generic->LDS pointers on
    // AMDGPU carry the AS(3) offset in the low 32 bits.
    const unsigned ldsAddr = (unsigned)(uintptr_t)(&row[0]) + voff;

    // CDNA5 async copy: global (SADDR base + per-lane VGPR offset) -> LDS,
    // 128 bits per lane, tracked with ASYNCcnt. Then drain our wave's async
    // counter before the workgroup barrier.
    asm volatile(
        "global_load_async_to_lds_b128 %0, %1, %2\n\t"
        "s_wait_asynccnt 0"
        :
        : "v"(ldsAddr), "v"(voff), "s"(gsrc)
        : "memory");

    __syncthreads();   // all waves drained their asynccnt -> full row in LDS

    // Pull this lane's 4 floats back (ds_load_b128); kept in registers for the
    // scale pass so HBM sees the input exactly once.
    v4f v = *reinterpret_cast<const v4f*>(&row[t * 4]);
    float partial = (v.x + v.y) + (v.z + v.w);

    // wave32 butterfly reduction via ds_swizzle XOR patterns (no address
    // math / bounds cndmask like __shfl_xor's ds_bpermute lowering).
    partial = swizzle_xor_add(partial, 16);
    partial = swizzle_xor_add(partial, 8);
    partial = swizzle_xor_add(partial, 4);
    partial = swizzle_xor_add(partial, 2);
    partial = swizzle_xor_add(partial, 1);

    if ((t & 31) == 0) wsum[t >> 5] = partial;
    __syncthreads();

    float deg = 0.0f;
    #pragma unroll
    for (int i = 0; i < NWAVES; ++i) deg += wsum[i];

    float inv = 1.0f / deg;
    // remove_nan_inf semantics: non-finite inverse -> 0 (comparison is false
    // for NaN too).
    inv = (__builtin_fabsf(inv) < __builtin_inff()) ? inv : 0.0f;

    v4f o;
    o.x = v.x * inv;
    o.y = v.y * inv;
    o.z = v.z * inv;
    o.w = v.w * inv;

    // Streaming output with no reuse: non-temporal store so the 256 MB output
    // stream doesn't evict the input stream from L2.
    __builtin_nontemporal_store(o, reinterpret_cast<v4f*>(out + rowBase + t * 4));
}

extern "C" void kernel_launch(void* const* d_in, const int* in_sizes, int n_in,
                              void* d_out, int out_size, void* d_ws, size_t ws_size,
                              hipStream_t stream) {
    const float* in = (const float*)d_in[0];
    float* out = (float*)d_out;

    const long long total = (long long)in_sizes[0];      // 2*32*1024*1024
    const int nrows = (int)(total / NCOLS);              // 65536 rows

    if (nrows > 0) {
        rownorm_kernel<<<dim3((unsigned)nrows), dim3(TPB), 0, stream>>>(in, out);
    }
}